// MacroToMesoEncoderTrafficFluid_30236569764426
// MI455X (gfx1250) — compile-verified
//
#include <hip/hip_runtime.h>
#include <math.h>

#define NCELLS 65536
#define QPTS   512
#define QF     512.0f
#define HSTEP  (70.0f / 511.0f)

typedef float v2f __attribute__((ext_vector_type(2)));
typedef float v8f __attribute__((ext_vector_type(8)));

// ---------------------------------------------------------------------------
// Kernel 1: per-cell Newton solve for lambda using closed-form geometric
// moments of the uniform-grid exponential family. xi_q = q*h, h = 70/511.
// With t = lambda*h and r = e^t:
//   S0 = (r^Q - 1)/(r - 1)
//   mean(q-units)  m(t)   = Q/(1 - e^{-Qt}) - 1/(1 - e^{-t})
//   var (q-units)  var(t) = e^{-t}/(1-e^{-t})^2 - Q^2 e^{-Qt}/(1-e^{-Qt})^2
// Both forms are overflow-free for either sign via |t| + symmetry
// m(-t) = (Q-1) - m(t); var is even. Taylor branch covers t -> 0 (the
// Newton starting point): m ~ (Q-1)/2 + t(Q^2-1)/12, var ~ (Q^2-1)/12.
// ---------------------------------------------------------------------------
__global__ void solve_lambda_mu(const float* __restrict__ macro_v,
                                float* __restrict__ lam_out,
                                float* __restrict__ mu0_out) {
  int i = blockIdx.x * blockDim.x + threadIdx.x;
  if (i >= NCELLS) return;

  const float mq_target = macro_v[i] / HSTEP;   // target mean in q-units
  float t = 0.0f;

  #pragma unroll 4
  for (int it = 0; it < 32; ++it) {
    float m, var;
    float s = fabsf(t);
    if (s < 1e-4f) {
      m   = 0.5f * (QF - 1.0f) + t * (QF * QF - 1.0f) * (1.0f / 12.0f);
      var = (QF * QF - 1.0f) * (1.0f / 12.0f);
    } else {
      float es = __expf(-s);
      float eQ = __expf(-QF * s);
      float d1 = 1.0f - es;
      float dQ = 1.0f - eQ;
      float mpos = QF / dQ - 1.0f / d1;
      m   = (t > 0.0f) ? mpos : (QF - 1.0f - mpos);
      var = es / (d1 * d1) - (QF * QF) * eQ / (dQ * dQ);
    }
    t = t - (m - mq_target) / var;
    t = fminf(0.3f, fmaxf(-0.3f, t));   // Newton never needs more; safety clamp
  }

  // mu0 = -log S0(t)
  float s = fabsf(t);
  float logS0;
  if (s < 1e-4f) {
    logS0 = __logf(QF) + 0.5f * (QF - 1.0f) * t
          + (QF * QF - 1.0f) * (1.0f / 24.0f) * t * t;
  } else {
    logS0 = ((t > 0.0f) ? (QF - 1.0f) * t : 0.0f)
          + log1pf(-__expf(-QF * s)) - log1pf(-__expf(-s));
  }

  lam_out[i] = t / HSTEP;
  mu0_out[i] = -logS0;
}

// ---------------------------------------------------------------------------
// Kernel 2: f_eq[n,q] = exp(lam_n * xi_q + mu0_n) for a 16x16 tile per wave.
// Exponent computed as a 16x16x4 f32 WMMA:  A = [lam | mu0 | 0 | 0] (16x4),
// B = [xi ; 1 ; 0 ; 0] (4x16), C = 0.  Then v_exp_f32 on the 8 accumulator
// VGPRs and coalesced b32 stores (lanes 0..15 of each VGPR cover 64
// contiguous bytes of one row; lanes 16..31 the row 8 below).
// A layout (32-bit 16x4): lanes 0-15 -> M=lane, VGPR0=K0, VGPR1=K1;
//                         lanes 16-31 -> K2,K3 (zero padding here).
// ---------------------------------------------------------------------------
__global__ void feq_outer_exp_wmma(const float* __restrict__ lam,
                                   const float* __restrict__ mu0,
                                   const float* __restrict__ xi,
                                   float* __restrict__ out) {
  const int wave = (blockIdx.x * blockDim.x + threadIdx.x) >> 5;
  const int lane = threadIdx.x & 31;

  const int TQ = QPTS / 16;            // 32 tiles along q
  const int tq = wave % TQ;
  const int tn = wave / TQ;
  const int q0 = tq * 16;
  const int n0 = tn * 16;

  const int  lo  = lane & 15;
  const bool low = (lane < 16);

  v2f a, b;
  a.x = low ? lam[n0 + lo] : 0.0f;     // K=0 row of A : lambda
  a.y = low ? mu0[n0 + lo] : 0.0f;     // K=1 row of A : mu0
  b.x = low ? xi[q0 + lo]  : 0.0f;     // K=0 row of B : xi
  b.y = low ? 1.0f         : 0.0f;     // K=1 row of B : ones

  v8f c = {};
  c = __builtin_amdgcn_wmma_f32_16x16x4_f32(
      /*neg_a=*/false, a, /*neg_b=*/false, b,
      /*c_mod=*/(short)0, c, /*reuse_a=*/false, /*reuse_b=*/false);

  const int mbase = (lane >> 4) * 8;   // lanes 16-31 hold rows M+8
  #pragma unroll
  for (int j = 0; j < 8; ++j) {
    float e = __expf(c[j]);            // v_exp_f32
    out[(size_t)(n0 + mbase + j) * QPTS + (q0 + lo)] = e;
  }
}

// ---------------------------------------------------------------------------
extern "C" void kernel_launch(void* const* d_in, const int* in_sizes, int n_in,
                              void* d_out, int out_size, void* d_ws, size_t ws_size,
                              hipStream_t stream) {
  const float* macro_v = (const float*)d_in[0];   // [N,1] f32
  const float* xi      = (const float*)d_in[1];   // [Q]   f32
  float*       out     = (float*)d_out;           // [N,Q] f32

  float* lam = (float*)d_ws;                      // N floats
  float* mu0 = lam + NCELLS;                      // N floats (512 KiB total)

  // Solver: one thread per cell.
  solve_lambda_mu<<<NCELLS / 256, 256, 0, stream>>>(macro_v, lam, mu0);

  // Output: one wave per 16x16 tile; 8 waves per 256-thread block.
  const int tiles  = (NCELLS / 16) * (QPTS / 16); // 131072 waves
  const int blocks = tiles / 8;                   // 16384 blocks
  feq_outer_exp_wmma<<<blocks, 256, 0, stream>>>(lam, mu0, xi, out);
}